// SpeGSA_32126355374629
// MI455X (gfx1250) — compile-verified
//
#include <hip/hip_runtime.h>

typedef __attribute__((ext_vector_type(2))) float v2f;
typedef __attribute__((ext_vector_type(8))) float v8f;

#define BATCH 8
#define CIN   192
#define G     64
#define H     256
#define W     256
#define HW    (H*W)           // 65536
#define OCH   256
#define NT    64              // pixels per workgroup in kernel B
#define VS    72              // LDS row stride (72 % 64 == 8 -> conflict-free B-frag reads, 16B aligned)
#define EPS   1e-5f

// ---------------------------------------------------------------------------
// Kernel A: pooled attention statistics -> per-(b,g) softmax weights (3)
// One workgroup (256 threads) per (b,g). Reads its 3 channels exactly once.
// ---------------------------------------------------------------------------
__global__ void __launch_bounds__(256)
attn_stats_kernel(const float* __restrict__ x,
                  const float* __restrict__ aw,     // (G,3)
                  float* __restrict__ attn_out)     // (B*G,3)
{
    const int bg = blockIdx.x;            // 0..511
    const int b  = bg >> 6;
    const int g  = bg & 63;
    const float* xc = x + ((size_t)b * CIN + (size_t)g * 3) * HW;

    __shared__ float s9[9];
    if (threadIdx.x < 9) s9[threadIdx.x] = 0.0f;
    __syncthreads();

    float acc[9];
#pragma unroll
    for (int t = 0; t < 9; ++t) acc[t] = 0.0f;

    // 4096 pooled windows, 16 per thread; consecutive lanes -> consecutive pw
    for (int n = threadIdx.x; n < 4096; n += 256) {
        const int ph = n >> 6;
        const int pw = n & 63;
        float q[3], k[3];
#pragma unroll
        for (int c3 = 0; c3 < 3; ++c3) {
            const float* base = xc + (size_t)c3 * HW + (size_t)(ph * 4) * W + pw * 4;
            float mx = -__builtin_inff();
            float sm = 0.0f;
#pragma unroll
            for (int r = 0; r < 4; ++r) {
                const float4 v = *reinterpret_cast<const float4*>(base + (size_t)r * W);
                mx = fmaxf(mx, fmaxf(fmaxf(v.x, v.y), fmaxf(v.z, v.w)));
                sm += (v.x + v.y) + (v.z + v.w);
            }
            q[c3] = mx;
            k[c3] = sm * (1.0f / 16.0f);
        }
#pragma unroll
        for (int i = 0; i < 3; ++i)
#pragma unroll
            for (int j = 0; j < 3; ++j)
                acc[i * 3 + j] += q[i] * k[j];
    }

    // wave32 butterfly reduce, then one atomic per wave per cell
#pragma unroll
    for (int t = 0; t < 9; ++t) {
        float v = acc[t];
#pragma unroll
        for (int off = 16; off >= 1; off >>= 1)
            v += __shfl_xor(v, off, 32);
        if ((threadIdx.x & 31) == 0) atomicAdd(&s9[t], v);
    }
    __syncthreads();

    if (threadIdx.x == 0) {
        const float sc = 4.0f / (float)H;     // 1/64
        const float w0 = aw[g * 3 + 0], w1 = aw[g * 3 + 1], w2 = aw[g * 3 + 2];
        float aj[3];
#pragma unroll
        for (int j = 0; j < 3; ++j)
            aj[j] = (s9[0 * 3 + j] * w0 + s9[1 * 3 + j] * w1 + s9[2 * 3 + j] * w2) * sc;
        const float m  = fmaxf(aj[0], fmaxf(aj[1], aj[2]));
        const float e0 = __expf(aj[0] - m);
        const float e1 = __expf(aj[1] - m);
        const float e2 = __expf(aj[2] - m);
        const float inv = 1.0f / (e0 + e1 + e2);
        attn_out[bg * 3 + 0] = e0 * inv;
        attn_out[bg * 3 + 1] = e1 * inv;
        attn_out[bg * 3 + 2] = e2 * inv;
    }
}

// ---------------------------------------------------------------------------
// Kernel B: fused  (attn-weighted channel mix -> BN1+ReLU) -> 1x1 conv GEMM
// via V_WMMA_F32_16X16X4_F32 -> BN2 -> store.
// Grid: (HW/NT, BATCH). 256 threads = 8 waves.
// M = 256 out-channels (16 m-tiles), K = 64 groups, N = 64 pixels (4 n-tiles).
// Wave w: n-tile = w&3, m-tiles = { 2*t + (w>>2) : t in 0..7 }.
// ---------------------------------------------------------------------------
__global__ void __launch_bounds__(256)
fused_mix_gemm_kernel(const float* __restrict__ x,
                      const float* __restrict__ attnw,   // (B*G,3) from kernel A
                      const float* __restrict__ bn1g, const float* __restrict__ bn1b,
                      const float* __restrict__ bn1m, const float* __restrict__ bn1v,
                      const float* __restrict__ w2,      // (OCH, G) row-major
                      const float* __restrict__ bn2g, const float* __restrict__ bn2b,
                      const float* __restrict__ bn2m, const float* __restrict__ bn2v,
                      float* __restrict__ out)
{
    __shared__ float V[G * VS];         // padded K x N staging (18 KB)
    __shared__ float GP[G * 4];         // per-g: a0*s1, a1*s1, a2*s1, bias1
    __shared__ float OP[OCH * 2];       // per-o: scale2, bias2

    const int b   = blockIdx.y;
    const int n0  = blockIdx.x * NT;
    const int tid = threadIdx.x;

    // fold BN1 into the 3 attention multipliers + one bias, per group
    if (tid < G) {
        const int gg = tid;
        const float s1 = bn1g[gg] * rsqrtf(bn1v[gg] + EPS);
        const float bb = bn1b[gg] - bn1m[gg] * s1;
        const float* a = attnw + ((size_t)b * G + gg) * 3;
        GP[gg * 4 + 0] = a[0] * s1;
        GP[gg * 4 + 1] = a[1] * s1;
        GP[gg * 4 + 2] = a[2] * s1;
        GP[gg * 4 + 3] = bb;
    }
    {   // fold BN2, per out-channel
        const int o  = tid;
        const float s2 = bn2g[o] * rsqrtf(bn2v[o] + EPS);
        OP[o * 2 + 0] = s2;
        OP[o * 2 + 1] = bn2b[o] - bn2m[o] * s2;
    }
    __syncthreads();

    // build V[g][n] = relu( c0*x0 + c1*x1 + c2*x2 + bias )
    // float4 per thread: 1024 units of 16B, fully coalesced, b128 LDS stores
    const float* xb = x + (size_t)b * CIN * HW + n0;
#pragma unroll
    for (int it = 0; it < (G * NT / 4) / 256; ++it) {
        const int unit = it * 256 + tid;
        const int gg = unit >> 4;           // group row
        const int q  = unit & 15;           // float4 index within row
        const float* p = xb + (size_t)(gg * 3) * HW + q * 4;
        const float4 x0 = *reinterpret_cast<const float4*>(p);
        const float4 x1 = *reinterpret_cast<const float4*>(p + HW);
        const float4 x2 = *reinterpret_cast<const float4*>(p + 2 * (size_t)HW);
        const float c0 = GP[gg * 4 + 0], c1 = GP[gg * 4 + 1];
        const float c2 = GP[gg * 4 + 2], bb = GP[gg * 4 + 3];
        float4 v;
        v.x = fmaxf(c0 * x0.x + c1 * x1.x + c2 * x2.x + bb, 0.0f);
        v.y = fmaxf(c0 * x0.y + c1 * x1.y + c2 * x2.y + bb, 0.0f);
        v.z = fmaxf(c0 * x0.z + c1 * x1.z + c2 * x2.z + bb, 0.0f);
        v.w = fmaxf(c0 * x0.w + c1 * x1.w + c2 * x2.w + bb, 0.0f);
        *reinterpret_cast<float4*>(&V[gg * VS + q * 4]) = v;
    }
    __syncthreads();

    const int wave = tid >> 5;
    const int lane = tid & 31;
    const int nt   = wave & 3;     // n-tile of this wave
    const int mgrp = wave >> 2;    // m-tile parity
    const int lh   = lane >> 4;    // selects K pair (and M+8 half of C/D)
    const int ll   = lane & 15;

    v8f acc[8];
#pragma unroll
    for (int t = 0; t < 8; ++t) acc[t] = (v8f){0,0,0,0,0,0,0,0};

    for (int kk = 0; kk < G / 4; ++kk) {
        const int kbase = 4 * kk + 2 * lh;
        // B fragment: B[k][n], n = nt*16 + ll, k = kbase + e  (conflict-free: VS=72)
        v2f bfrag;
        bfrag.x = V[(kbase + 0) * VS + nt * 16 + ll];
        bfrag.y = V[(kbase + 1) * VS + nt * 16 + ll];
#pragma unroll
        for (int t = 0; t < 8; ++t) {
            const int mt = t * 2 + mgrp;
            // A fragment: A[m][k] = w2[m*G + k], m = mt*16 + ll (8B-aligned float2, L2/WGP$ resident)
            const float2 wv = *reinterpret_cast<const float2*>(w2 + (size_t)(mt * 16 + ll) * G + kbase);
            v2f afrag;
            afrag.x = wv.x;
            afrag.y = wv.y;
            acc[t] = __builtin_amdgcn_wmma_f32_16x16x4_f32(
                false, afrag, false, bfrag, (short)0, acc[t], false, false);
        }
    }

    // epilogue: BN2 + coalesced store. VGPR r: lanes0-15 -> M=r, lanes16-31 -> M=r+8
    float* ob = out + (size_t)b * OCH * HW + n0 + nt * 16 + ll;
#pragma unroll
    for (int t = 0; t < 8; ++t) {
        const int mt = t * 2 + mgrp;
#pragma unroll
        for (int r = 0; r < 8; ++r) {
            const int o = mt * 16 + r + lh * 8;
            ob[(size_t)o * HW] = acc[t][r] * OP[o * 2 + 0] + OP[o * 2 + 1];
        }
    }
}

// ---------------------------------------------------------------------------
extern "C" void kernel_launch(void* const* d_in, const int* in_sizes, int n_in,
                              void* d_out, int out_size, void* d_ws, size_t ws_size,
                              hipStream_t stream) {
    const float* x    = (const float*)d_in[0];
    const float* aw   = (const float*)d_in[1];
    const float* bn1g = (const float*)d_in[2];
    const float* bn1b = (const float*)d_in[3];
    const float* bn1m = (const float*)d_in[4];
    const float* bn1v = (const float*)d_in[5];
    const float* w2   = (const float*)d_in[6];
    const float* bn2g = (const float*)d_in[7];
    const float* bn2b = (const float*)d_in[8];
    const float* bn2m = (const float*)d_in[9];
    const float* bn2v = (const float*)d_in[10];
    float* out   = (float*)d_out;
    float* attnw = (float*)d_ws;   // BATCH*G*3 floats = 6 KB

    attn_stats_kernel<<<dim3(BATCH * G), dim3(256), 0, stream>>>(x, aw, attnw);

    fused_mix_gemm_kernel<<<dim3(HW / NT, BATCH), dim3(256), 0, stream>>>(
        x, attnw, bn1g, bn1b, bn1m, bn1v, w2, bn2g, bn2b, bn2m, bn2v, out);
}